// GCN_34729105555796
// MI455X (gfx1250) — compile-verified
//
#include <hip/hip_runtime.h>
#include <hip/hip_bf16.h>

typedef __attribute__((ext_vector_type(2))) float v2f;
typedef __attribute__((ext_vector_type(8))) float v8f;

#define N_NODES 50000
#define N_EDGES 800000
#define CH      64
#define N_GRAPHS 256
#define BN_EPS  1e-5f

// ---------------- degree / normalization ----------------
__global__ void k_deg_init(float* __restrict__ deg) {
    int i = blockIdx.x * blockDim.x + threadIdx.x;
    if (i < N_NODES) deg[i] = 1.0f;            // self-loop contribution
}

__global__ void k_deg_count(const int* __restrict__ dst, float* __restrict__ deg) {
    int e = blockIdx.x * blockDim.x + threadIdx.x;
    if (e < N_EDGES) atomicAdd(&deg[dst[e]], 1.0f);
}

__global__ void k_deg_rsqrt(float* __restrict__ deg) {
    int i = blockIdx.x * blockDim.x + threadIdx.x;
    if (i < N_NODES) deg[i] = rsqrtf(deg[i]);  // deg >= 1 always (self loops)
}

// ---------------- WMMA f32 matmul: hw'[i,:] = dis[i] * (h @ W)[i,:] ----------
// One wave computes a 16x64 tile via V_WMMA_F32_16X16X4_F32 (exact f32 path).
// W staged in LDS as float2 pairs Wp[kk*64+n] = {W[2kk][n], W[2kk+1][n]} so a
// B fragment (VGPR0=K{0,2}, VGPR1=K{1,3} per ISA layout) is one 64-bit load.
__global__ __launch_bounds__(256)
void k_matmul(const float* __restrict__ h, const float* __restrict__ W,
              const float* __restrict__ dis, float* __restrict__ hw,
              float* __restrict__ agg) {
    __shared__ float2 Wp[32 * 64];
    int tid = threadIdx.x;
    for (int e = tid; e < 32 * 64; e += 256) {
        int kk = e >> 6, n = e & 63;
        Wp[e] = float2{W[(2 * kk) * CH + n], W[(2 * kk + 1) * CH + n]};
    }
    __syncthreads();

    const int wave = tid >> 5;
    const int lane = tid & 31;
    const int tile = blockIdx.x * 8 + wave;         // wave-uniform
    if (tile >= N_NODES / 16) return;               // uniform exit: EXEC stays all-1s
    const int row0 = tile * 16;
    const int lr = lane & 15;
    const int hi = lane >> 4;                       // 0: K+0/K+1, 1: K+2/K+3

    v8f acc0 = {}, acc1 = {}, acc2 = {}, acc3 = {};
    const float* arow = h + (size_t)(row0 + lr) * CH;

    for (int k = 0; k < CH; k += 4) {
        // A fragment: lane<16 -> M=lr K={k,k+1}; lane>=16 -> M=lr K={k+2,k+3}
        v2f a = *(const v2f*)(arow + k + 2 * hi);
        int kb = (k >> 1) + hi;
        v2f b0 = *(const v2f*)&Wp[kb * 64 + 0 * 16 + lr];
        v2f b1 = *(const v2f*)&Wp[kb * 64 + 1 * 16 + lr];
        v2f b2 = *(const v2f*)&Wp[kb * 64 + 2 * 16 + lr];
        v2f b3 = *(const v2f*)&Wp[kb * 64 + 3 * 16 + lr];
        acc0 = __builtin_amdgcn_wmma_f32_16x16x4_f32(false, a, false, b0, (short)0, acc0, false, false);
        acc1 = __builtin_amdgcn_wmma_f32_16x16x4_f32(false, a, false, b1, (short)0, acc1, false, false);
        acc2 = __builtin_amdgcn_wmma_f32_16x16x4_f32(false, a, false, b2, (short)0, acc2, false, false);
        acc3 = __builtin_amdgcn_wmma_f32_16x16x4_f32(false, a, false, b3, (short)0, acc3, false, false);
    }

    // C/D layout: VGPR r -> M=r (lanes 0-15) / M=r+8 (lanes 16-31), N=lr
    float dv[8];
#pragma unroll
    for (int r = 0; r < 8; ++r) dv[r] = dis[row0 + hi * 8 + r];
#pragma unroll
    for (int r = 0; r < 8; ++r) {
        size_t base = (size_t)(row0 + hi * 8 + r) * CH + lr;
        float v0 = acc0[r] * dv[r], v1 = acc1[r] * dv[r];
        float v2 = acc2[r] * dv[r], v3 = acc3[r] * dv[r];
        hw[base]       = v0; hw[base + 16]  = v1; hw[base + 32] = v2; hw[base + 48] = v3;
        agg[base]      = v0; agg[base + 16] = v1; agg[base + 32] = v2; agg[base + 48] = v3;  // self-loop init
    }
}

// ---------------- edge scatter: agg[dst,:] += hw'[src,:] --------------------
// 32 lanes per edge, float2 gather (L2-resident), 2 non-returning f32 atomics.
__global__ __launch_bounds__(256)
void k_scatter(const int* __restrict__ ei, const float* __restrict__ hw,
               float* __restrict__ agg) {
    int t = blockIdx.x * blockDim.x + threadIdx.x;
    int e = t >> 5;
    int lane = t & 31;
    if (e >= N_EDGES) return;
    int s = ei[e];
    int d = ei[N_EDGES + e];
    float2 v = *(const float2*)(hw + (size_t)s * CH + lane * 2);
    float* o = agg + (size_t)d * CH + lane * 2;
    atomicAdd(o, v.x);
    atomicAdd(o + 1, v.y);
}

// ---------------- finalize conv + BN statistics -----------------------------
__global__ void k_zero_bn(float* __restrict__ a, float* __restrict__ b) {
    int t = threadIdx.x;
    if (t < CH) { a[t] = 0.0f; b[t] = 0.0f; }
}

__global__ __launch_bounds__(256)
void k_finalize(float* __restrict__ agg, const float* __restrict__ dis,
                const float* __restrict__ bias, float* __restrict__ bnsum,
                float* __restrict__ bnsumsq) {
    int gt = blockIdx.x * blockDim.x + threadIdx.x;
    int c = gt & 63;
    int grp = gt >> 6;
    int ngrp = (gridDim.x * blockDim.x) >> 6;
    float b = bias[c];
    float s = 0.0f, s2 = 0.0f;
    for (int row = grp; row < N_NODES; row += ngrp) {
        size_t idx = (size_t)row * CH + c;
        float v = dis[row] * agg[idx] + b;
        agg[idx] = v;
        s += v; s2 += v * v;
    }
    atomicAdd(&bnsum[c], s);
    atomicAdd(&bnsumsq[c], s2);
}

__global__ void k_bncoef(const float* __restrict__ bnsum, const float* __restrict__ bnsumsq,
                         const float* __restrict__ gamma, const float* __restrict__ beta,
                         float* __restrict__ coef) {
    int c = threadIdx.x;
    if (c < CH) {
        float mean = bnsum[c] * (1.0f / N_NODES);
        float var = bnsumsq[c] * (1.0f / N_NODES) - mean * mean;
        float a = gamma[c] * rsqrtf(var + BN_EPS);
        coef[c] = a;
        coef[CH + c] = beta[c] - a * mean;
    }
}

__global__ __launch_bounds__(256)
void k_bnrelu(const float* __restrict__ agg, const float* __restrict__ coef,
              float* __restrict__ h) {
    int t = blockIdx.x * blockDim.x + threadIdx.x;
    if (t < N_NODES * CH) {
        int c = t & 63;
        float v = coef[c] * agg[t] + coef[CH + c];
        h[t] = v > 0.0f ? v : 0.0f;
    }
}

// ---------------- pooling + MLP head ----------------------------------------
__global__ void k_pool_zero(float* __restrict__ pool, float* __restrict__ cnt) {
    int t = blockIdx.x * blockDim.x + threadIdx.x;
    if (t < N_GRAPHS * CH) pool[t] = 0.0f;
    if (t < N_GRAPHS) cnt[t] = 0.0f;
}

__global__ __launch_bounds__(256)
void k_pool(const float* __restrict__ h, const int* __restrict__ batch,
            float* __restrict__ pool, float* __restrict__ cnt) {
    int t = blockIdx.x * blockDim.x + threadIdx.x;
    if (t >= N_NODES * CH) return;
    int i = t >> 6, c = t & 63;
    int g = batch[i];
    atomicAdd(&pool[g * CH + c], h[t]);
    if (c == 0) atomicAdd(&cnt[g], 1.0f);
}

__global__ void k_fc1(const float* __restrict__ pool, const float* __restrict__ cnt,
                      const float* __restrict__ w, const float* __restrict__ b,
                      float* __restrict__ out1) {
    int t = blockIdx.x * blockDim.x + threadIdx.x;
    if (t >= N_GRAPHS * CH) return;
    int g = t >> 6, j = t & 63;
    float inv = 1.0f / fmaxf(cnt[g], 1.0f);
    float acc = b[j];
    for (int k = 0; k < CH; ++k) acc += pool[g * CH + k] * inv * w[k * CH + j];
    out1[t] = fmaxf(acc, 0.0f);
}

__global__ void k_fc2(const float* __restrict__ out1, const float* __restrict__ w,
                      const float* __restrict__ b, float* __restrict__ out) {
    int t = blockIdx.x * blockDim.x + threadIdx.x;
    if (t >= N_GRAPHS * 10) return;
    int g = t / 10, j = t % 10;
    float acc = b[j];
    for (int k = 0; k < CH; ++k) acc += out1[g * CH + k] * w[k * 10 + j];
    out[t] = acc;
}

// ---------------- host ------------------------------------------------------
extern "C" void kernel_launch(void* const* d_in, const int* in_sizes, int n_in,
                              void* d_out, int out_size, void* d_ws, size_t ws_size,
                              hipStream_t stream) {
    (void)in_sizes; (void)n_in; (void)out_size; (void)ws_size;
    const float* x     = (const float*)d_in[0];
    const int*   ei    = (const int*)d_in[1];
    const int*   batch = (const int*)d_in[2];
    const float* W1    = (const float*)d_in[3];
    const float* b1    = (const float*)d_in[4];
    const float* Wh    = (const float*)d_in[5];
    const float* bh    = (const float*)d_in[6];
    const float* gamma = (const float*)d_in[7];
    const float* beta  = (const float*)d_in[8];
    const float* fc1_w = (const float*)d_in[9];
    const float* fc1_b = (const float*)d_in[10];
    const float* fc2_w = (const float*)d_in[11];
    const float* fc2_b = (const float*)d_in[12];
    float* out = (float*)d_out;

    char* ws = (char*)d_ws;
    size_t off = 0;
    auto alloc = [&](size_t bytes) -> void* {
        void* p = ws + off;
        off += (bytes + 255) & ~(size_t)255;
        return p;
    };
    float* dis     = (float*)alloc((size_t)N_NODES * 4);
    float* hw      = (float*)alloc((size_t)N_NODES * CH * 4);
    float* agg     = (float*)alloc((size_t)N_NODES * CH * 4);
    float* h       = (float*)alloc((size_t)N_NODES * CH * 4);
    float* bnsum   = (float*)alloc(CH * 4);
    float* bnsumsq = (float*)alloc(CH * 4);
    float* coef    = (float*)alloc(2 * CH * 4);
    float* pool    = (float*)alloc((size_t)N_GRAPHS * CH * 4);
    float* cnt     = (float*)alloc((size_t)N_GRAPHS * 4);
    float* o1      = (float*)alloc((size_t)N_GRAPHS * CH * 4);

    k_deg_init<<<(N_NODES + 255) / 256, 256, 0, stream>>>(dis);
    k_deg_count<<<(N_EDGES + 255) / 256, 256, 0, stream>>>(ei + N_EDGES, dis);
    k_deg_rsqrt<<<(N_NODES + 255) / 256, 256, 0, stream>>>(dis);

    const int mmBlocks = (N_NODES / 16 + 7) / 8;           // 391 blocks, 8 waves each
    const int scBlocks = (N_EDGES * 32) / 256;             // 100000
    const int ewBlocks = (N_NODES * CH + 255) / 256;

    for (int l = 0; l < 4; ++l) {
        const float* hin = (l == 0) ? x : h;
        const float* W   = (l == 0) ? W1 : Wh + (size_t)(l - 1) * CH * CH;
        const float* bb  = (l == 0) ? b1 : bh + (size_t)(l - 1) * CH;
        k_zero_bn<<<1, 64, 0, stream>>>(bnsum, bnsumsq);
        k_matmul<<<mmBlocks, 256, 0, stream>>>(hin, W, dis, hw, agg);
        k_scatter<<<scBlocks, 256, 0, stream>>>(ei, hw, agg);
        k_finalize<<<128, 256, 0, stream>>>(agg, dis, bb, bnsum, bnsumsq);
        k_bncoef<<<1, 64, 0, stream>>>(bnsum, bnsumsq, gamma + l * CH, beta + l * CH, coef);
        k_bnrelu<<<ewBlocks, 256, 0, stream>>>(agg, coef, h);
    }

    k_pool_zero<<<(N_GRAPHS * CH + 255) / 256, 256, 0, stream>>>(pool, cnt);
    k_pool<<<ewBlocks, 256, 0, stream>>>(h, batch, pool, cnt);
    k_fc1<<<(N_GRAPHS * CH + 255) / 256, 256, 0, stream>>>(pool, cnt, fc1_w, fc1_b, o1);
    k_fc2<<<(N_GRAPHS * 10 + 255) / 256, 256, 0, stream>>>(o1, fc2_w, fc2_b, out);
}